// TaskMoE_36730560315505
// MI455X (gfx1250) — compile-verified
//
#include <hip/hip_runtime.h>

// Problem constants (match reference)
#define NTOK 4096      // B*S = 8*512
#define DDIM 1024
#define HDIM 2048
#define NE   16
#define TOPK 2
#define TM   16                       // token tile (one WMMA M)
#define PCAP (NTOK*TOPK + NE*TM)      // 8448: pairs + per-expert padding
#define PTILES (PCAP/TM)              // 528

typedef float v2f __attribute__((ext_vector_type(2)));
typedef float v8f __attribute__((ext_vector_type(8)));

__device__ __forceinline__ float gelu_tanh(float v) {
    const float c = 0.7978845608028654f;
    float u = c * (v + 0.044715f * v * v * v);
    return 0.5f * v * (1.0f + tanhf(u));
}

// ---------------- K0: init ----------------
__global__ void moe_k0_init(float* __restrict__ out, int* __restrict__ pair_token,
                            int* __restrict__ counts) {
    int i = blockIdx.x * blockDim.x + threadIdx.x;
    const int tot4 = NTOK * DDIM / 4;
    if (i < tot4) {
        float4 z = {0.f, 0.f, 0.f, 0.f};
        ((float4*)out)[i] = z;
    }
    if (i < PCAP) pair_token[i] = -1;
    if (i < NE) counts[i] = 0;
}

// ---------------- K1: gating (softmax + top-2), one wave per token ----------------
__global__ void moe_k1_gate(const float* __restrict__ x, const float* __restrict__ wg,
                            int* __restrict__ counts, int* __restrict__ tok_e,
                            float* __restrict__ tok_g) {
    const int t = blockIdx.x * 4 + (threadIdx.x >> 5);
    const int lane = threadIdx.x & 31;
    const int e = lane & 15;
    const int half = lane >> 4;
    const float* xr = x + (size_t)t * DDIM;

    float acc = 0.f;
    for (int d0 = 0; d0 < DDIM; d0 += 2) {
        int d = d0 + half;
        acc = fmaf(xr[d], wg[d * NE + e], acc);
    }
    acc += __shfl_xor(acc, 16, 32);   // logit for expert e (duplicated on both halves)

    // softmax over 16 experts (butterfly within 16-lane group)
    float m = acc;
    for (int s = 8; s >= 1; s >>= 1) m = fmaxf(m, __shfl_xor(m, s, 32));
    float p = __expf(acc - m);
    float sum = p;
    for (int s = 8; s >= 1; s >>= 1) sum += __shfl_xor(sum, s, 32);
    float prob = p / sum;

    // top-1 (ties -> lower index, as jax top_k)
    float v1 = prob; int i1 = e;
    for (int s = 8; s >= 1; s >>= 1) {
        float ov = __shfl_xor(v1, s, 32);
        int   oi = __shfl_xor(i1, s, 32);
        if (ov > v1 || (ov == v1 && oi < i1)) { v1 = ov; i1 = oi; }
    }
    // top-2
    float v2 = (e == i1) ? -1.f : prob; int i2 = e;
    for (int s = 8; s >= 1; s >>= 1) {
        float ov = __shfl_xor(v2, s, 32);
        int   oi = __shfl_xor(i2, s, 32);
        if (ov > v2 || (ov == v2 && oi < i2)) { v2 = ov; i2 = oi; }
    }
    if (lane == 0) {
        tok_e[2 * t]     = i1;  tok_g[2 * t]     = v1;
        tok_e[2 * t + 1] = i2;  tok_g[2 * t + 1] = v2;
        atomicAdd(&counts[i1], 1);
        atomicAdd(&counts[i2], 1);
    }
}

// ---------------- K2: scan offsets (aligned to TM) + slot->expert map ----------------
__global__ void moe_k2_scan(const int* __restrict__ counts, int* __restrict__ offsets,
                            int* __restrict__ cursors, int* __restrict__ slot_expert) {
    __shared__ int soff[NE + 1];
    if (threadIdx.x == 0) {
        int off = 0;
        for (int e = 0; e < NE; ++e) {
            soff[e] = off; offsets[e] = off; cursors[e] = off;
            off += (counts[e] + TM - 1) & ~(TM - 1);
        }
        soff[NE] = off; offsets[NE] = off;
    }
    __syncthreads();
    for (int s = threadIdx.x; s < PCAP; s += blockDim.x) {
        int e = 0;
        for (int i = 0; i < NE; ++i)
            if (s >= soff[i] && s < soff[i + 1]) e = i;
        slot_expert[s] = (s < soff[NE]) ? e : 0;
    }
}

// ---------------- K3: scatter tokens into expert-grouped slots ----------------
__global__ void moe_k3_scatter(const int* __restrict__ tok_e, const float* __restrict__ tok_g,
                               int* __restrict__ cursors, int* __restrict__ pair_token,
                               float* __restrict__ pair_gate) {
    int t = blockIdx.x * blockDim.x + threadIdx.x;
    if (t >= NTOK) return;
    #pragma unroll
    for (int k = 0; k < TOPK; ++k) {
        int e = tok_e[2 * t + k];
        int slot = atomicAdd(&cursors[e], 1);
        pair_token[slot] = t;
        pair_gate[slot] = tok_g[2 * t + k];
    }
}

// ---------------- K4: h = gelu(x @ w1[e] + b1[e]) via V_WMMA_F32_16X16X4_F32 ------
// block: 16 slots x 256 H-cols, 128 threads (4 waves x 4 WMMA col tiles)
__global__ void moe_k4_ffn1(const float* __restrict__ x, const float* __restrict__ w1,
                            const float* __restrict__ b1, const int* __restrict__ pair_token,
                            const int* __restrict__ slot_expert, float* __restrict__ hbuf) {
    extern __shared__ float smem[];               // 16 x 1024 floats = 64 KB
    const int slot0 = blockIdx.x * TM;
    const int colBase = blockIdx.y * 256;
    const int tid = threadIdx.x;

    // Stage x rows for the 16 slots (zeros for pad slots)
    for (int i = tid; i < TM * (DDIM / 4); i += 128) {
        int row = i >> 8;            // / 256
        int c4  = i & 255;
        int tok = pair_token[slot0 + row];
        float4 v = {0.f, 0.f, 0.f, 0.f};
        if (tok >= 0) v = ((const float4*)(x + (size_t)tok * DDIM))[c4];
        ((float4*)smem)[row * (DDIM / 4) + c4] = v;
    }
    __syncthreads();

    const int e = slot_expert[slot0];
    const float* w1e = w1 + (size_t)e * DDIM * HDIM;
    const float* b1e = b1 + (size_t)e * HDIM;

    const int lane = tid & 31;
    const int wv   = tid >> 5;
    const int l15  = lane & 15;
    const int half = lane >> 4;
    const int j0   = colBase + wv * 64;

    v8f acc[4] = {};
    for (int dk = 0; dk < DDIM; dk += 4) {
        int ak = dk + 2 * half;                       // K rows {0,1} / {2,3} per half-wave
        v2f a = *(const v2f*)(smem + l15 * DDIM + ak);
        #pragma unroll
        for (int n = 0; n < 4; ++n) {
            int col = j0 + n * 16 + l15;
            v2f b;
            b.x = w1e[(size_t)ak * HDIM + col];
            b.y = w1e[(size_t)(ak + 1) * HDIM + col];
            acc[n] = __builtin_amdgcn_wmma_f32_16x16x4_f32(
                false, a, false, b, (short)0, acc[n], false, false);
        }
    }

    #pragma unroll
    for (int n = 0; n < 4; ++n) {
        int col = j0 + n * 16 + l15;
        float bias = b1e[col];
        #pragma unroll
        for (int i = 0; i < 8; ++i) {
            int mrow = i + 8 * half;                  // C layout: VGPR i -> M=i / M=i+8
            hbuf[(size_t)(slot0 + mrow) * HDIM + col] = gelu_tanh(acc[n][i] + bias);
        }
    }
}

// ---------------- K5: y += gate * (h @ w2[e] + b2[e]) --------------------------
// block: 16 slots x 256 D-cols, 128 threads (4 waves x 4 WMMA col tiles)
__global__ void moe_k5_ffn2(const float* __restrict__ w2, const float* __restrict__ b2,
                            const int* __restrict__ pair_token, const float* __restrict__ pair_gate,
                            const int* __restrict__ slot_expert, const float* __restrict__ hbuf,
                            float* __restrict__ out) {
    extern __shared__ float smem[];               // 16 x 2048 floats = 128 KB
    const int slot0 = blockIdx.x * TM;
    const int colBase = blockIdx.y * 256;
    const int tid = threadIdx.x;

    for (int i = tid; i < TM * (HDIM / 4); i += 128) {
        int row = i >> 9;            // / 512
        int c4  = i & 511;
        ((float4*)smem)[row * (HDIM / 4) + c4] =
            ((const float4*)(hbuf + (size_t)(slot0 + row) * HDIM))[c4];
    }
    __syncthreads();

    const int e = slot_expert[slot0];
    const float* w2e = w2 + (size_t)e * HDIM * DDIM;
    const float* b2e = b2 + (size_t)e * DDIM;

    const int lane = tid & 31;
    const int wv   = tid >> 5;
    const int l15  = lane & 15;
    const int half = lane >> 4;
    const int j0   = colBase + wv * 64;

    v8f acc[4] = {};
    for (int hk = 0; hk < HDIM; hk += 4) {
        int ak = hk + 2 * half;
        v2f a = *(const v2f*)(smem + l15 * HDIM + ak);
        #pragma unroll
        for (int n = 0; n < 4; ++n) {
            int col = j0 + n * 16 + l15;
            v2f b;
            b.x = w2e[(size_t)ak * DDIM + col];
            b.y = w2e[(size_t)(ak + 1) * DDIM + col];
            acc[n] = __builtin_amdgcn_wmma_f32_16x16x4_f32(
                false, a, false, b, (short)0, acc[n], false, false);
        }
    }

    #pragma unroll
    for (int n = 0; n < 4; ++n) {
        int col = j0 + n * 16 + l15;
        float bias = b2e[col];
        #pragma unroll
        for (int i = 0; i < 8; ++i) {
            int mrow = i + 8 * half;
            int tok = pair_token[slot0 + mrow];
            if (tok >= 0) {
                float g = pair_gate[slot0 + mrow];
                unsafeAtomicAdd(out + (size_t)tok * DDIM + col, g * (acc[n][i] + bias));
            }
        }
    }
}

extern "C" void kernel_launch(void* const* d_in, const int* in_sizes, int n_in,
                              void* d_out, int out_size, void* d_ws, size_t ws_size,
                              hipStream_t stream) {
    const float* x      = (const float*)d_in[0];
    const float* w_gate = (const float*)d_in[1];
    const float* w1     = (const float*)d_in[2];
    const float* b1     = (const float*)d_in[3];
    const float* w2     = (const float*)d_in[4];
    const float* b2     = (const float*)d_in[5];
    float* out = (float*)d_out;

    char* ws = (char*)d_ws;
    size_t off = 0;
    float* hbuf       = (float*)(ws + off); off += (size_t)PCAP * HDIM * sizeof(float);
    int*   pair_token = (int*)  (ws + off); off += PCAP * sizeof(int);
    float* pair_gate  = (float*)(ws + off); off += PCAP * sizeof(float);
    int*   slot_exp   = (int*)  (ws + off); off += PCAP * sizeof(int);
    int*   tok_e      = (int*)  (ws + off); off += NTOK * TOPK * sizeof(int);
    float* tok_g      = (float*)(ws + off); off += NTOK * TOPK * sizeof(float);
    int*   counts     = (int*)  (ws + off); off += 32 * sizeof(int);
    int*   offsets    = (int*)  (ws + off); off += 32 * sizeof(int);
    int*   cursors    = (int*)  (ws + off); off += 32 * sizeof(int);

    // K0: zero out + routing init
    {
        int tot4 = NTOK * DDIM / 4;
        int blocks = (tot4 + 255) / 256;
        moe_k0_init<<<blocks, 256, 0, stream>>>(out, pair_token, counts);
    }
    // K1: gating (one wave per token)
    moe_k1_gate<<<NTOK / 4, 128, 0, stream>>>(x, w_gate, counts, tok_e, tok_g);
    // K2: offsets / expert map
    moe_k2_scan<<<1, 256, 0, stream>>>(counts, offsets, cursors, slot_exp);
    // K3: scatter
    moe_k3_scatter<<<NTOK / 256, 256, 0, stream>>>(tok_e, tok_g, cursors, pair_token, pair_gate);
    // K4: first grouped GEMM + gelu
    {
        dim3 grid(PTILES, HDIM / 256);
        moe_k4_ffn1<<<grid, 128, TM * DDIM * sizeof(float), stream>>>(
            x, w1, b1, pair_token, slot_exp, hbuf);
    }
    // K5: second grouped GEMM + gated combine
    {
        dim3 grid(PTILES, DDIM / 256);
        moe_k5_ffn2<<<grid, 128, TM * HDIM * sizeof(float), stream>>>(
            w2, b2, pair_token, pair_gate, slot_exp, hbuf, out);
    }
}